// DST_1133871366817
// MI455X (gfx1250) — compile-verified
//
#include <hip/hip_runtime.h>

// DST-II of shape (4096, 8192): Y = X * S^T, S[k][n] = sin(pi/N (n+1/2)(k+1)).
// Tiled fp32 WMMA GEMM. Sine basis generated exactly in LDS (integer range
// reduction mod 4N = 2^15 + v_sin_f32 in revolutions). A tile staged with
// CDNA5 async global->LDS loads (ASYNCcnt); both LDS tiles stored K-contiguous
// so every fragment is one conflict-free ds_load_b64.

typedef float v2f __attribute__((ext_vector_type(2)));
typedef float v8f __attribute__((ext_vector_type(8)));

#define N_COLS  8192
#define M_BLK   128       // rows of X per block
#define N_BLK   128       // output columns per block
#define KB      32        // K (reduction) chunk staged in LDS
#define LDK     (KB + 4)  // 36: 16B-aligned rows, conflict-free frag loads
#define THREADS 256       // 8 wave32

__global__ __launch_bounds__(THREADS)
void dst2_wmma_kernel(const float* __restrict__ x, float* __restrict__ y) {
    __shared__ float sA[M_BLK * LDK];   // X tile,   [M][K] K-contiguous
    __shared__ float sB[N_BLK * LDK];   // sine tile,[N][K] K-contiguous

    const int tid     = threadIdx.x;
    const int lane    = tid & 31;
    const int wave    = tid >> 5;
    const int wm      = wave & 3;       // 4 M-groups of 32 rows
    const int wn      = wave >> 2;      // 2 N-groups of 64 cols
    const int m0      = blockIdx.y * M_BLK;
    const int n0      = blockIdx.x * N_BLK;
    const int halfSel = lane >> 4;      // 0 = lanes 0-15, 1 = lanes 16-31
    const int lrow    = lane & 15;

    v8f acc[2][4];
    const v8f zero = {0.f, 0.f, 0.f, 0.f, 0.f, 0.f, 0.f, 0.f};
#pragma unroll
    for (int i = 0; i < 2; ++i)
#pragma unroll
        for (int j = 0; j < 4; ++j)
            acc[i][j] = zero;

    // B-generation coords: each thread owns 16 K-contiguous sines of one column
    const int ncol = tid >> 1;          // 0..127 (output column within block)
    const int kh   = (tid & 1) * 16;    // 0 or 16
    const unsigned colp1 = (unsigned)(n0 + ncol + 1);   // (k+1) in DST formula

    for (int kg = 0; kg < N_COLS; kg += KB) {
        // ---- stage X tile: async global -> LDS, 4 x b128 per thread ----
#pragma unroll
        for (int i = 0; i < 4; ++i) {
            const int f4 = tid + THREADS * i;   // 0..1023 float4 slots
            const int r  = f4 >> 3;             // 8 float4 per KB=32 row
            const int c4 = f4 & 7;
            const float*   gptr = x + (size_t)(m0 + r) * N_COLS + kg + c4 * 4;
            const unsigned ldst = (unsigned)(size_t)(&sA[r * LDK + c4 * 4]);
            asm volatile("global_load_async_to_lds_b128 %0, %1, off"
                         :: "v"(ldst), "v"(gptr) : "memory");
        }

        // ---- generate sine basis tile (transposed, K-contiguous) ----
        // sB[n][k] = sin(2*pi * ((2*(kg+k)+1)*(n0+n+1) mod 2^15) / 2^15)
        {
            float vals[16];
#pragma unroll
            for (int j = 0; j < 16; ++j) {
                const unsigned twoKp1 = 2u * (unsigned)(kg + kh + j) + 1u;
                const unsigned ip     = (twoKp1 * colp1) & 32767u;  // exact mod 4N
                vals[j] = __builtin_amdgcn_sinf((float)ip * (1.0f / 32768.0f));
            }
#pragma unroll
            for (int q = 0; q < 4; ++q)
                *reinterpret_cast<float4*>(&sB[ncol * LDK + kh + q * 4]) =
                    make_float4(vals[q * 4], vals[q * 4 + 1],
                                vals[q * 4 + 2], vals[q * 4 + 3]);
        }
        asm volatile("s_wait_asynccnt 0x0" ::: "memory");
        __syncthreads();

        // ---- MMA over the chunk: 8 K-steps of 4, 2x4 tiles per wave ----
#pragma unroll
        for (int kk = 0; kk < KB; kk += 4) {
            const int kb = kk + halfSel * 2;    // this lane's even K pair
            v2f afrag[2];
#pragma unroll
            for (int mi = 0; mi < 2; ++mi) {
                const int r = wm * 32 + mi * 16 + lrow;
                afrag[mi] = *reinterpret_cast<const v2f*>(&sA[r * LDK + kb]);
            }
            v2f bfrag[4];
#pragma unroll
            for (int ni = 0; ni < 4; ++ni) {
                const int c = wn * 64 + ni * 16 + lrow;
                bfrag[ni] = *reinterpret_cast<const v2f*>(&sB[c * LDK + kb]);
            }
#pragma unroll
            for (int mi = 0; mi < 2; ++mi)
#pragma unroll
                for (int ni = 0; ni < 4; ++ni)
                    acc[mi][ni] = __builtin_amdgcn_wmma_f32_16x16x4_f32(
                        false, afrag[mi], false, bfrag[ni],
                        (short)0, acc[mi][ni], false, false);
        }
        __syncthreads();
    }

    // ---- epilogue: C/D layout = VGPR r -> M = r (lanes 0-15) / r+8 (16-31) ----
#pragma unroll
    for (int mi = 0; mi < 2; ++mi) {
#pragma unroll
        for (int ni = 0; ni < 4; ++ni) {
            const int colg = n0 + wn * 64 + ni * 16 + lrow;
#pragma unroll
            for (int r = 0; r < 8; ++r) {
                const int rowg = m0 + wm * 32 + mi * 16 + r + halfSel * 8;
                y[(size_t)rowg * N_COLS + colg] = acc[mi][ni][r];
            }
        }
    }
}

extern "C" void kernel_launch(void* const* d_in, const int* in_sizes, int n_in,
                              void* d_out, int out_size, void* d_ws, size_t ws_size,
                              hipStream_t stream) {
    (void)n_in; (void)d_ws; (void)ws_size; (void)out_size;
    const float* x = (const float*)d_in[0];
    float*       y = (float*)d_out;
    const int rows = in_sizes[0] / N_COLS;          // 4096
    dim3 grid(N_COLS / N_BLK, rows / M_BLK);        // (64, 32)
    dim3 block(THREADS);
    hipLaunchKernelGGL(dst2_wmma_kernel, grid, block, 0, stream, x, y);
}